// MultiBoxLoss_49581102465797
// MI455X (gfx1250) — compile-verified
//
#include <hip/hip_runtime.h>
#include <stdint.h>

#define B_ 32
#define P_ 32768
#define C_ 81
#define NEGPOS 3
#define SEL_THREADS 1024

typedef __attribute__((ext_vector_type(4))) unsigned int u32x4_t;
typedef __attribute__((ext_vector_type(8))) int i32x8_t;
typedef __attribute__((ext_vector_type(4))) int i32x4_t;

// ---------------------------------------------------------------------------
// Kernel 1: per-prior CE (single pass over conf_data, logits cached in VGPRs),
// smooth-L1 for positives, ce_mine written to workspace, deterministic
// per-block partial sums, integer pos counts per batch row.
// ---------------------------------------------------------------------------
__global__ __launch_bounds__(256) void mb_ce_kernel(
    const float* __restrict__ loc_data, const float* __restrict__ conf_data,
    const float* __restrict__ loc_t, const int* __restrict__ conf_t,
    float* __restrict__ ce_mine, int* __restrict__ numpos,
    float* __restrict__ partial_l, float* __restrict__ partial_posce)
{
    __shared__ float red_l[256];
    __shared__ float red_c[256];
    __shared__ int   red_n;
    const int tid = threadIdx.x;
    if (tid == 0) red_n = 0;
    __syncthreads();

    const size_t i = (size_t)blockIdx.x * 256 + tid;   // prior index in [0, B*P)
    const int t  = conf_t[i];
    const bool pos = t > 0;

    // ---- localization smooth-L1 (16B-aligned float4 loads) ----
    const float4 a = ((const float4*)loc_data)[i];
    const float4 b = ((const float4*)loc_t)[i];
    float dl = 0.f, dx;
    dx = fabsf(a.x - b.x); dl += (dx < 1.f) ? 0.5f * dx * dx : dx - 0.5f;
    dx = fabsf(a.y - b.y); dl += (dx < 1.f) ? 0.5f * dx * dx : dx - 0.5f;
    dx = fabsf(a.z - b.z); dl += (dx < 1.f) ? 0.5f * dx * dx : dx - 0.5f;
    dx = fabsf(a.w - b.w); dl += (dx < 1.f) ? 0.5f * dx * dx : dx - 0.5f;
    dl = pos ? dl : 0.f;

    // ---- per-prior softmax CE: cache 81 logits in registers, one HBM pass ----
    const float* c = conf_data + i * (size_t)C_;
    float v[C_];
#pragma unroll
    for (int j = 0; j < C_; ++j) v[j] = c[j];
    float m = v[0];
#pragma unroll
    for (int j = 1; j < C_; ++j) m = fmaxf(m, v[j]);
    float s = 0.f;
#pragma unroll
    for (int j = 0; j < C_; ++j) s += __expf(v[j] - m);
    const float ce = m + __logf(s) - c[t];   // lse - true_logit  (>= 0)

    ce_mine[i] = pos ? 0.f : ce;
    const float cpos = pos ? ce : 0.f;

    if (pos) atomicAdd(&red_n, 1);           // integer: deterministic
    red_l[tid] = dl;
    red_c[tid] = cpos;
    __syncthreads();
    for (int st = 128; st > 0; st >>= 1) {   // fixed-order tree reduction
        if (tid < st) { red_l[tid] += red_l[tid + st]; red_c[tid] += red_c[tid + st]; }
        __syncthreads();
    }
    if (tid == 0) {
        partial_l[blockIdx.x]     = red_l[0];
        partial_posce[blockIdx.x] = red_c[0];
        atomicAdd(&numpos[blockIdx.x >> 7], red_n);  // 128 blocks per batch row
    }
}

// ---------------------------------------------------------------------------
// Kernel 2: one workgroup per batch row.  Stage the 131 KB ce_mine row into
// LDS via the Tensor Data Mover (CDNA5 async DMA), then exact k-th-largest
// radix select (4 x 256-bin rounds; ce >= 0 so float bits are u32-monotone)
// and threshold-based sum:  sum_topk = sum_{v > v*} v + rem * v*.
// ---------------------------------------------------------------------------
__global__ __launch_bounds__(SEL_THREADS) void mb_select_kernel(
    const float* __restrict__ ce_mine, const int* __restrict__ numpos,
    float* __restrict__ negsum)
{
    __shared__ float    srow[P_];        // 128 KB: whole row (CDNA5 320 KB LDS)
    __shared__ unsigned hist[256];
    __shared__ unsigned sh_pref;
    __shared__ unsigned sh_mask;
    __shared__ int      sh_rem;
    __shared__ float    red[SEL_THREADS];

    const int row = blockIdx.x;
    const int tid = threadIdx.x;
    const float* gr = ce_mine + (size_t)row * P_;

    const int np = numpos[row];
    int k = NEGPOS * np;
    if (k > P_ - 1) k = P_ - 1;
    if (k <= 0) { if (tid == 0) negsum[row] = 0.f; return; }   // uniform exit

#if __has_builtin(__builtin_amdgcn_tensor_load_to_lds) && __has_builtin(__builtin_amdgcn_s_wait_tensorcnt)
    if (tid == 0) {
        const unsigned lds_off = (unsigned)(size_t)&srow[0];
        const unsigned long long ga = (unsigned long long)(size_t)gr;
        u32x4_t g0; i32x8_t g1;
        i32x4_t g2 = {0, 0, 0, 0};
        i32x4_t g3 = {0, 0, 0, 0};
        i32x8_t g4 = {0, 0, 0, 0, 0, 0, 0, 0};
        // D# group 0: count=1 | lds_addr | global_addr[56:0] | type=2
        g0[0] = 1u;
        g0[1] = lds_off;
        g0[2] = (unsigned)ga;
        g0[3] = (unsigned)((ga >> 32) & 0x1FFFFFFu) | (2u << 30);
        // D# group 1: data_size=4B, tensor_dim0 = tile_dim0 = P, dim1 = 1
        g1[0] = (int)(2u << 16);                              // data_size=2 (4B)
        g1[1] = (int)(((unsigned)P_ & 0xFFFFu) << 16);        // tensor_dim0 lo16
        g1[2] = (int)(((unsigned)P_ >> 16) | (1u << 16));     // dim0 hi16, tensor_dim1=1
        g1[3] = (int)(((unsigned)P_ & 0xFFFFu) << 16);        // tile_dim0 = P
        g1[4] = 1;                                            // tile_dim1 = 1
        g1[5] = (int)P_;                                      // tensor_dim0_stride
        g1[6] = 0;
        g1[7] = 0;
        // clang-23 / therock-10.0 form: 6 args (g0, g1, g2, g3, g4, cpol)
        __builtin_amdgcn_tensor_load_to_lds(g0, g1, g2, g3, g4, 0);
        __builtin_amdgcn_s_wait_tensorcnt(0);                 // issuing wave waits
    }
#else
    for (int e = tid; e < P_; e += SEL_THREADS) srow[e] = gr[e];
#endif
    __syncthreads();

    if (tid == 0) { sh_pref = 0u; sh_mask = 0u; sh_rem = k; }
    __syncthreads();

    for (int shift = 24; shift >= 0; shift -= 8) {
        if (tid < 256) hist[tid] = 0u;
        __syncthreads();
        const unsigned pref = sh_pref, mask = sh_mask;
        for (int e = tid; e < P_; e += SEL_THREADS) {
            const unsigned u = __float_as_uint(srow[e]);
            if ((u & mask) == pref) atomicAdd(&hist[(u >> shift) & 255u], 1u);
        }
        __syncthreads();
        if (tid == 0) {
            const int rem = sh_rem;
            unsigned cum = 0; int bsel = 0;
            for (int bin = 255; bin >= 0; --bin) {
                const unsigned cc = hist[bin];
                if (cum + cc >= (unsigned)rem) { bsel = bin; break; }
                cum += cc;
            }
            sh_rem  = rem - (int)cum;
            sh_pref = pref | ((unsigned)bsel << shift);
            sh_mask = mask | (0xFFu << shift);
        }
        __syncthreads();
    }

    const unsigned thr = sh_pref;   // bit pattern of k-th largest value v*
    const int rem = sh_rem;         // # of elements equal to v* that are selected
    float part = 0.f;
    for (int e = tid; e < P_; e += SEL_THREADS) {
        const float x = srow[e];
        part += (__float_as_uint(x) > thr) ? x : 0.f;
    }
    red[tid] = part;
    __syncthreads();
    for (int st = SEL_THREADS / 2; st > 0; st >>= 1) {
        if (tid < st) red[tid] += red[tid + st];
        __syncthreads();
    }
    if (tid == 0) negsum[row] = red[0] + (float)rem * __uint_as_float(thr);
}

// ---------------------------------------------------------------------------
// Kernel 3: deterministic final reduction -> two scalar outputs.
// ---------------------------------------------------------------------------
__global__ __launch_bounds__(256) void mb_final_kernel(
    const float* __restrict__ partial_l, const float* __restrict__ partial_posce,
    const int* __restrict__ numpos, const float* __restrict__ negsum,
    float* __restrict__ out, int nblk)
{
    __shared__ float rl[256], rc[256];
    const int tid = threadIdx.x;
    float sl = 0.f, sc = 0.f;
    for (int i = tid; i < nblk; i += 256) { sl += partial_l[i]; sc += partial_posce[i]; }
    rl[tid] = sl; rc[tid] = sc;
    __syncthreads();
    for (int st = 128; st > 0; st >>= 1) {
        if (tid < st) { rl[tid] += rl[tid + st]; rc[tid] += rc[tid + st]; }
        __syncthreads();
    }
    if (tid == 0) {
        int tp = 0; float ns = 0.f;
        for (int r = 0; r < B_; ++r) { tp += numpos[r]; ns += negsum[r]; }
        float N = (float)tp; if (N < 1.f) N = 1.f;
        out[0] = rl[0] / N;
        out[1] = (rc[0] + ns) / N;
    }
}

// ---------------------------------------------------------------------------
extern "C" void kernel_launch(void* const* d_in, const int* in_sizes, int n_in,
                              void* d_out, int out_size, void* d_ws, size_t ws_size,
                              hipStream_t stream)
{
    const float* loc_data  = (const float*)d_in[0];
    const float* conf_data = (const float*)d_in[1];
    const float* loc_t     = (const float*)d_in[2];
    const int*   conf_t    = (const int*)d_in[3];
    float* out = (float*)d_out;

    char* ws = (char*)d_ws;
    float* ce_mine = (float*)ws;                                  // B*P floats (4 MB)
    size_t off = (size_t)B_ * P_ * sizeof(float);
    int*   numpos = (int*)(ws + off);   off += 128;               // 32 ints
    float* negsum = (float*)(ws + off); off += 128;               // 32 floats
    const int nblk = (B_ * P_) / 256;                             // 4096
    float* partial_l     = (float*)(ws + off); off += (size_t)nblk * sizeof(float);
    float* partial_posce = (float*)(ws + off);

    (void)hipMemsetAsync(numpos, 0, B_ * sizeof(int), stream);

    mb_ce_kernel<<<nblk, 256, 0, stream>>>(loc_data, conf_data, loc_t, conf_t,
                                           ce_mine, numpos, partial_l, partial_posce);
    mb_select_kernel<<<B_, SEL_THREADS, 0, stream>>>(ce_mine, numpos, negsum);
    mb_final_kernel<<<1, 256, 0, stream>>>(partial_l, partial_posce, numpos, negsum,
                                           out, nblk);
}